// SpectoCapsLayer_3298534884325
// MI455X (gfx1250) — compile-verified
//
#include <hip/hip_runtime.h>

typedef __attribute__((ext_vector_type(2))) float v2f;
typedef __attribute__((ext_vector_type(8))) float v8f;

#define B_DIM 256
#define I_DIM 1152
#define O_DIM 10
#define F_DIM 8
#define G_DIM 16

// Batch-row stride inside u_dash [B, I, O, G], in floats.
// j * BSTRIDE * 4B = j * 737280 B  (j<8 -> max 5.16 MB, fits signed 24-bit
// IOFFSET) so all 8 row-stores share one vaddr + immediate offsets.
static constexpr size_t BSTRIDE = (size_t)I_DIM * O_DIM * G_DIM;

// ---------------------------------------------------------------------------
// Pass 1: u_dash[b,i,o,g] = sum_f x[b,i,f] * W[i,o,f,g]
// Per (i,o) this is a GEMM: M=B=256, K=F=8, N=G=16 -> 16 M-tiles of
// V_WMMA_F32_16X16X4_F32 (two chained K=4 steps). fp32-exact matrix path.
// Grid: (I, O); block: 256 threads = 8 waves; each wave owns 2 M-tiles.
// ---------------------------------------------------------------------------
__global__ __launch_bounds__(256) void udash_wmma_kernel(
    const float* __restrict__ x,      // [B, I, F]
    const float* __restrict__ W,      // [I, O, F, G]
    float* __restrict__ u_out)        // [B, I, O, G]
{
  const int i    = blockIdx.x;
  const int o    = blockIdx.y;
  const int lane = threadIdx.x & 31;
  const int wave = threadIdx.x >> 5;
  const int n    = lane & 15;          // N index (g), lanes 0-15 / 16-31
  const bool hi  = lane >= 16;
  const int k0   = hi ? 2 : 0;         // K pair held by this lane half

  // B matrix (4x16 per WMMA), shared by all M-tiles in the block.
  // VGPR0 = {K=k0 row}, VGPR1 = {K=k0+1 row}, lanes carry N.
  const float* Wp = W + (size_t)(i * O_DIM + o) * (F_DIM * G_DIM);
  v2f bk0, bk1;
  bk0.x = Wp[(k0 + 0) * G_DIM + n];
  bk0.y = Wp[(k0 + 1) * G_DIM + n];
  bk1.x = Wp[(k0 + 4) * G_DIM + n];
  bk1.y = Wp[(k0 + 5) * G_DIM + n];

  // Uniform (i,o) part of the output address; per-lane adds n and mbase.
  float* const ubase = u_out + ((size_t)i * O_DIM + o) * G_DIM + n;

#pragma unroll
  for (int t = 0; t < 2; ++t) {
    const int b_base = (wave * 2 + t) * 16;
    const int brow   = b_base + n;     // A-matrix row for this lane
    const float* xp  = x + ((size_t)brow * I_DIM + i) * F_DIM;
    const float4 xlo = *(const float4*)(xp);       // f = 0..3
    const float4 xh4 = *(const float4*)(xp + 4);   // f = 4..7
    // A 16x4 layout: lanes = M; VGPR0 = {K0 | K2}, VGPR1 = {K1 | K3}
    v2f a0, a1;
    a0.x = hi ? xlo.z : xlo.x;
    a0.y = hi ? xlo.w : xlo.y;
    a1.x = hi ? xh4.z : xh4.x;
    a1.y = hi ? xh4.w : xh4.y;

    v8f c = {};
    c = __builtin_amdgcn_wmma_f32_16x16x4_f32(false, a0, false, bk0,
                                              (short)0, c, false, false);
    c = __builtin_amdgcn_wmma_f32_16x16x4_f32(false, a1, false, bk1,
                                              (short)0, c, false, false);

    // D layout: VGPR j = row M=j (lanes 0-15) / M=j+8 (lanes 16-31), N=lane%16
    const int mbase = b_base + (hi ? 8 : 0);
    float* const up = ubase + (size_t)mbase * BSTRIDE;  // one vaddr per tile
#pragma unroll
    for (int j = 0; j < 8; ++j)
      up[(size_t)j * BSTRIDE] = c[j];  // immediate offsets j*737280 B
  }
}

// ---------------------------------------------------------------------------
// Pass 2: dynamic routing, fully fused per (b,o). Each block keeps its
// u[b,:,o,:] slice (1152x16 fp32) in registers (5 rows x 16 per thread) and
// runs all 3 routing iterations; LDS only for I-dimension reductions.
// Grid: (B, O); block: 256 threads.
// ---------------------------------------------------------------------------
__global__ __launch_bounds__(256) void routing_kernel(
    const float* __restrict__ u_dash,  // [B, I, O, G]
    float* __restrict__ v_out)         // [B, O, G]
{
  __shared__ float red[256];
  __shared__ float vecred[256 * 17];   // padded stride to dodge bank conflicts
  __shared__ float sjv[G_DIM];

  const int b   = blockIdx.x;
  const int o   = blockIdx.y;
  const int tid = threadIdx.x;

  float ureg[5][G_DIM];
  float lg[5];

  // Load this block's slice: thread owns i = tid, tid+256, ... (4 or 5 rows)
#pragma unroll
  for (int k = 0; k < 5; ++k) {
    const int i = k * 256 + tid;
    lg[k] = 0.0f;
    if (i < I_DIM) {
      const float* up = u_dash + (((size_t)b * I_DIM + i) * O_DIM + o) * G_DIM;
#pragma unroll
      for (int g4 = 0; g4 < 4; ++g4) {
        const float4 v = *(const float4*)(up + g4 * 4);
        ureg[k][g4 * 4 + 0] = v.x;
        ureg[k][g4 * 4 + 1] = v.y;
        ureg[k][g4 * 4 + 2] = v.z;
        ureg[k][g4 * 4 + 3] = v.w;
      }
    } else {
#pragma unroll
      for (int g = 0; g < G_DIM; ++g) ureg[k][g] = 0.0f;
    }
  }

  float scale = 0.0f;
  for (int step = 0; step < 3; ++step) {
    // ---- softmax over I: block max of logits
    float lmax = -3.4e38f;
#pragma unroll
    for (int k = 0; k < 5; ++k)
      if (k * 256 + tid < I_DIM) lmax = fmaxf(lmax, lg[k]);
    __syncthreads();
    red[tid] = lmax;
    __syncthreads();
    for (int s = 128; s > 0; s >>= 1) {
      if (tid < s) red[tid] = fmaxf(red[tid], red[tid + s]);
      __syncthreads();
    }
    const float m = red[0];
    __syncthreads();

    // ---- fused: sum of exp + partial s_j = sum_i exp(l_i - m) * u_i
    float lsum = 0.0f;
    float part[G_DIM];
#pragma unroll
    for (int g = 0; g < G_DIM; ++g) part[g] = 0.0f;
#pragma unroll
    for (int k = 0; k < 5; ++k) {
      if (k * 256 + tid < I_DIM) {
        const float e = __expf(lg[k] - m);
        lsum += e;
#pragma unroll
        for (int g = 0; g < G_DIM; ++g) part[g] += e * ureg[k][g];
      }
    }
    red[tid] = lsum;
#pragma unroll
    for (int g = 0; g < G_DIM; ++g) vecred[tid * 17 + g] = part[g];
    __syncthreads();
    for (int s = 128; s > 0; s >>= 1) {
      if (tid < s) red[tid] += red[tid + s];
      __syncthreads();
    }
    const float S = red[0];

    // ---- reduce the 16-wide partial sums; divide by softmax denominator
    if (tid < G_DIM) {
      float acc = 0.0f;
      for (int t2 = 0; t2 < 256; ++t2) acc += vecred[t2 * 17 + tid];
      sjv[tid] = acc / S;               // s_j[g]
    }
    __syncthreads();

    // ---- squash scale: |s| / (|s|^2 + 1)
    float n2 = 0.0f;
#pragma unroll
    for (int g = 0; g < G_DIM; ++g) { const float s = sjv[g]; n2 += s * s; }
    scale = sqrtf(n2) / (n2 + 1.0f);

    // ---- logits += <u_i, v_j>
#pragma unroll
    for (int k = 0; k < 5; ++k) {
      if (k * 256 + tid < I_DIM) {
        float d = 0.0f;
#pragma unroll
        for (int g = 0; g < G_DIM; ++g) d += ureg[k][g] * sjv[g];
        lg[k] += scale * d;
      }
    }
    __syncthreads();
  }

  if (tid < G_DIM)
    v_out[((size_t)b * O_DIM + o) * G_DIM + tid] = scale * sjv[tid];
}

// ---------------------------------------------------------------------------
extern "C" void kernel_launch(void* const* d_in, const int* in_sizes, int n_in,
                              void* d_out, int out_size, void* d_ws, size_t ws_size,
                              hipStream_t stream) {
  const float* x = (const float*)d_in[0];   // [256, 1152, 8]
  const float* W = (const float*)d_in[1];   // [1152, 10, 8, 16]
  float* v_out = (float*)d_out;             // [256, 10, 16]
  float* u_out = v_out + (size_t)B_DIM * O_DIM * G_DIM;  // [256,1152,10,16]

  dim3 g1(I_DIM, O_DIM);
  udash_wmma_kernel<<<g1, 256, 0, stream>>>(x, W, u_out);

  dim3 g2(B_DIM, O_DIM);
  routing_kernel<<<g2, 256, 0, stream>>>(u_out, v_out);
}